// GCNBench_72962904424515
// MI455X (gfx1250) — compile-verified
//
#include <hip/hip_runtime.h>

typedef __attribute__((ext_vector_type(2))) float v2f;
typedef __attribute__((ext_vector_type(8))) float v8f;

#define N_NODES 100000
#define FDIM 64   // HID == OUT == 64

// ---------------------------------------------------------------------------
// GEMM: C[M x 64] = A[M x K] * B[K x 64], f32, via V_WMMA_F32_16X16X4_F32.
// One wave -> 16 rows x 64 cols (4 N-tiles). Block = 256 threads = 8 waves
// = 128 rows. B staged through LDS in 64-row chunks (16 KB), stored in a
// fragment-native layout so each B fragment is ONE aligned ds_load_b64:
//
//   sB dword address of element (k, n), k in [0,64), n in [0,64):
//       t = k >> 1                (K-pair row, 0..31)
//       addr = t*128 + ((2n + (k&1)) ^ ((t&1) * 32))
//
// The two K values of a fragment (K = k0 + 2*half + {0,1}) are adjacent,
// and the XOR-32 swizzle makes lane-halves (t even vs t odd) hit disjoint
// bank groups -> conflict-free wave32 b64 reads.
//
// A-fragment (ISA 7.12.2, 32-bit A 16x4): lane half = lane>>4 picks K pair
// {0,1} vs {2,3}; VGPR j holds K = 2*half + j, row M = lane&15 -> one
// global_load_b64 per fragment. C/D: VGPR v, half h -> row v + 8*h.
// ---------------------------------------------------------------------------
template <int K>
__global__ __launch_bounds__(256) void gemm_wmma_kernel(
    const float* __restrict__ A, const float* __restrict__ Bg,
    float* __restrict__ C, int M)
{
    __shared__ float sB[64 * FDIM];   // 16 KB chunk of B (swizzled)

    const int wave = threadIdx.x >> 5;
    const int lane = threadIdx.x & 31;
    const int half = lane >> 4;       // 0 or 1
    const int ml   = lane & 15;
    const int row0 = (blockIdx.x * 8 + wave) * 16;
    const bool active = (row0 < M);

    // Inactive tail waves still run the barrier loop; point them at row 0.
    const float* arow = A + (size_t)((active ? row0 : 0) + ml) * K;

    v8f acc[4] = {};

    for (int kb = 0; kb < K; kb += 64) {
        __syncthreads();
        // Fill: each thread handles one (K-pair, n): two coalesced global
        // loads, one conflict-free ds_store_b64 into the swizzled layout.
        for (int idx = threadIdx.x; idx < 32 * FDIM; idx += 256) {
            const int t = idx >> 6;          // K-pair row 0..31
            const int n = idx & 63;          // column
            v2f bpair;
            bpair.x = Bg[(kb + 2 * t + 0) * FDIM + n];
            bpair.y = Bg[(kb + 2 * t + 1) * FDIM + n];
            *(v2f*)&sB[t * 128 + ((2 * n) ^ ((t & 1) * 32))] = bpair;
        }
        __syncthreads();

        if (kb + 64 < K)
            __builtin_prefetch(arow + kb + 64, 0, 0);   // global_prefetch_b8

#pragma unroll
        for (int k0 = 0; k0 < 64; k0 += 4) {
            // A fragment: two consecutive K values starting at k0 + 2*half
            v2f a = *(const v2f*)(arow + kb + k0 + 2 * half);
            const int t = (k0 >> 1) + half;  // this fragment's K-pair row
#pragma unroll
            for (int nt = 0; nt < 4; ++nt) {
                const v2f b = *(const v2f*)&sB[t * 128 +
                    ((2 * (nt * 16 + ml)) ^ ((t & 1) * 32))];
                acc[nt] = __builtin_amdgcn_wmma_f32_16x16x4_f32(
                    false, a, false, b, (short)0, acc[nt], false, false);
            }
        }
    }

    if (active) {
#pragma unroll
        for (int nt = 0; nt < 4; ++nt) {
#pragma unroll
            for (int v = 0; v < 8; ++v) {
                C[(size_t)(row0 + v + 8 * half) * FDIM + nt * 16 + ml] =
                    acc[nt][v];
            }
        }
    }
}

// ---------------------------------------------------------------------------
// SpMM (fixed degree 16): out[i,:] = sum_e vals[16i+e] * B[colind[16i+e], :]
// One wave per row. Lanes 0..15 preload the row's 16 (val, col) pairs; the
// unrolled edge loop broadcasts them with __shfl (v_readlane on wave32).
// Each lane owns 2 feature columns (float2 gather -> L2-resident B).
// ---------------------------------------------------------------------------
template <bool RELU>
__global__ __launch_bounds__(256) void spmm_deg16_kernel(
    const int* __restrict__ colind, const float* __restrict__ vals,
    const float* __restrict__ B, float* __restrict__ out, int M)
{
    const int wave = threadIdx.x >> 5;
    const int lane = threadIdx.x & 31;
    const int row  = blockIdx.x * 8 + wave;
    if (row >= M) return;

    const int base = row * 16;
    const int e    = lane & 15;
    const float vv = vals[base + e];
    const int   cc = colind[base + e];

    float ax = 0.0f, ay = 0.0f;
#pragma unroll
    for (int j = 0; j < 16; ++j) {
        const float v = __shfl(vv, j, 32);
        const int   c = __shfl(cc, j, 32);
        const float2 b = *(const float2*)(B + (size_t)c * FDIM + 2 * lane);
        ax = fmaf(v, b.x, ax);
        ay = fmaf(v, b.y, ay);
    }
    if (RELU) {
        ax = fmaxf(ax, 0.0f);
        ay = fmaxf(ay, 0.0f);
    }
    float2 r;
    r.x = ax;
    r.y = ay;
    *(float2*)(out + (size_t)row * FDIM + 2 * lane) = r;
}

// ---------------------------------------------------------------------------
// Pipeline: H = X@W1 ; G = relu(A@H) ; H = G@W2 ; out = A@H
// ---------------------------------------------------------------------------
extern "C" void kernel_launch(void* const* d_in, const int* in_sizes, int n_in,
                              void* d_out, int out_size, void* d_ws, size_t ws_size,
                              hipStream_t stream)
{
    const float* X      = (const float*)d_in[0];   // [N, 256]
    const float* W1     = (const float*)d_in[1];   // [256, 64]
    const float* W2     = (const float*)d_in[2];   // [64, 64]
    const float* vals   = (const float*)d_in[3];   // [N*16]
    // d_in[4] = rowptr: fixed-degree CSR (arange * 16), edges of row i are
    // exactly [16i, 16i+16) -> not needed on device.
    const int*   colind = (const int*)d_in[5];     // [N*16]
    float*       out    = (float*)d_out;           // [N, 64]

    float* H = (float*)d_ws;                       // N*64 f32
    float* G = H + (size_t)N_NODES * FDIM;         // N*64 f32

    const int M = N_NODES;
    dim3 blk(256);
    dim3 gemm_grid((M + 127) / 128);
    dim3 spmm_grid((M + 7) / 8);

    gemm_wmma_kernel<256><<<gemm_grid, blk, 0, stream>>>(X, W1, H, M);
    spmm_deg16_kernel<true><<<spmm_grid, blk, 0, stream>>>(colind, vals, H, G, M);
    gemm_wmma_kernel<64><<<gemm_grid, blk, 0, stream>>>(G, W2, H, M);
    spmm_deg16_kernel<false><<<spmm_grid, blk, 0, stream>>>(colind, vals, H, out, M);
}